// TemporalFeatureGraph_74955769249852
// MI455X (gfx1250) — compile-verified
//
#include <hip/hip_runtime.h>
#include <hip/hip_bf16.h>
#include <math.h>

typedef __attribute__((ext_vector_type(2))) float v2f;
typedef __attribute__((ext_vector_type(8))) float v8f;

#define B_    8
#define N_    256
#define W_    64
#define H_    64
#define PAIRS 65280   // 256*255
#define NEG_A 0.4f    // 0.4*att factor for |.| term
// out layout (floats): [0,522240) src idx, [522240,1044480) dst idx,
// [1044480,1566720) edge weights, [1566720, +33554432) alpha_seq
#define OUT_ALPHA_OFF 1566720

// ---------------------------------------------------------------------------
// Kernel 1: base[t][h] = cos(t*freqs) @ W[:,2:].T + bias  via WMMA f32 16x16x4
// 64x64 output = 4x4 tiles of 16x16, one wave each (16 waves = 512 threads).
// K=5 padded to 8 -> two k-steps of 4.
// ---------------------------------------------------------------------------
__global__ __launch_bounds__(512) void base_wmma_kernel(
    const float* __restrict__ Ww,   // (64,7) row-major
    const float* __restrict__ Wb,   // (64,)
    float* __restrict__ base)       // (64,64) [t][h]
{
  const int wave = threadIdx.x >> 5;   // 0..15
  const int lane = threadIdx.x & 31;
  const int tt   = wave >> 2;          // t-tile 0..3
  const int ht   = wave & 3;           // h-tile 0..3
  const int half = lane >> 4;          // 0: lanes 0-15, 1: lanes 16-31
  const int l    = lane & 15;

  // freqs = 10^-linspace(0,9,5)
  const float fr0 = 1.0f;
  const float fr1 = 5.623413251903491e-3f;   // 10^-2.25
  const float fr2 = 3.1622776601683794e-5f;  // 10^-4.5
  const float fr3 = 1.7782794100389228e-7f;  // 10^-6.75
  const float fr4 = 1e-9f;

  // A: time_emb tile, 16x4 f32: lanes 0-15 hold K=0,1 in v[0],v[1]; lanes 16-31 K=2,3
  const float t = (float)(tt * 16 + l);
  v2f a0, a1;
  if (half == 0) { a0.x = cosf(t * fr0); a0.y = cosf(t * fr1); }
  else           { a0.x = cosf(t * fr2); a0.y = cosf(t * fr3); }
  if (half == 0) { a1.x = cosf(t * fr4); a1.y = 0.0f; }
  else           { a1.x = 0.0f;          a1.y = 0.0f; }

  // B: 4x16 f32, B[k][n] = Ww[n][2+k]; lanes 0-15: K=0,1; lanes 16-31: K=2,3
  const int n = ht * 16 + l;
  v2f b0, b1;
  if (half == 0) { b0.x = Ww[n * 7 + 2]; b0.y = Ww[n * 7 + 3]; }
  else           { b0.x = Ww[n * 7 + 4]; b0.y = Ww[n * 7 + 5]; }
  if (half == 0) { b1.x = Ww[n * 7 + 6]; b1.y = 0.0f; }
  else           { b1.x = 0.0f;          b1.y = 0.0f; }

  // C: bias broadcast down each column n
  const float bias = Wb[n];
  v8f c;
#pragma unroll
  for (int r = 0; r < 8; ++r) c[r] = bias;

  c = __builtin_amdgcn_wmma_f32_16x16x4_f32(false, a0, false, b0, (short)0, c, false, false);
  c = __builtin_amdgcn_wmma_f32_16x16x4_f32(false, a1, false, b1, (short)0, c, false, false);

  // D 16x16 f32: VGPR r -> M = r + 8*half, N = l
#pragma unroll
  for (int r = 0; r < 8; ++r) {
    const int row = tt * 16 + r + 8 * half;
    base[row * 64 + ht * 16 + l] = c[r];
  }
}

// ---------------------------------------------------------------------------
// Kernel 2: attention + softmax.
// e[b,j,k,t] = sum_h att[h]*lrelu(hj*w1[h] + hk*w2[h] + base[t][h])
//            = 0.6*(hj*S1 + Cdot_k) + sum_h (0.4*att[h])*|hj*w1[h] + c_k[h]|
// Block = (j-tile of 16, t, b), 256 threads, thread = k. 2 FMA per inner h.
// ---------------------------------------------------------------------------
__global__ __launch_bounds__(256) void attn_kernel(
    const float* __restrict__ h,     // (8,256,64)
    const float* __restrict__ Ww,    // (64,7)
    const float* __restrict__ att,   // (64,)
    const float* __restrict__ base,  // (64,64) [t][h]
    float* __restrict__ alpha)       // (8,256,256,64) [b][j][k][t]
{
  const int jt = blockIdx.x;   // 0..15
  const int t  = blockIdx.y;   // 0..63
  const int b  = blockIdx.z;   // 0..7
  const int tid = threadIdx.x; // = k

  __shared__ float w1s[64], w2s[64], a4s[64], bts[64];
  __shared__ float et[16][257];   // e tile, padded

  if (tid < 64) {
    w1s[tid] = Ww[tid * 7 + 0];
    w2s[tid] = Ww[tid * 7 + 1];
    a4s[tid] = NEG_A * att[tid];
    bts[tid] = base[t * 64 + tid];
  }
  __syncthreads();

  const int k = tid;
  const float hk = h[((size_t)(b * N_ + k)) * W_ + t];

  // register caches
  float c[64], w1r[64], a4r[64];
  float Cdot = 0.0f, S1 = 0.0f;
#pragma unroll
  for (int hh = 0; hh < 64; ++hh) {
    const float w1v = w1s[hh];
    const float a4v = a4s[hh];
    const float cc  = fmaf(hk, w2s[hh], bts[hh]);
    w1r[hh] = w1v;
    a4r[hh] = a4v;
    c[hh]   = cc;
    const float attv = a4v * 2.5f;          // att[h]
    Cdot = fmaf(attv, cc, Cdot);            // sum att*(hk*w2+base)
    S1   = fmaf(attv, w1v, S1);             // sum att*w1
  }

  const float* __restrict__ hj_row = h + ((size_t)(b * N_ + jt * 16)) * W_ + t;
  __builtin_prefetch(hj_row, 0, 0);

  for (int jj = 0; jj < 16; ++jj) {
    const float hj = hj_row[(size_t)jj * W_];   // uniform -> scalar load
    float acc0 = 0.f, acc1 = 0.f, acc2 = 0.f, acc3 = 0.f;
#pragma unroll
    for (int hh = 0; hh < 64; hh += 4) {
      const float p0 = fmaf(hj, w1r[hh + 0], c[hh + 0]);
      const float p1 = fmaf(hj, w1r[hh + 1], c[hh + 1]);
      const float p2 = fmaf(hj, w1r[hh + 2], c[hh + 2]);
      const float p3 = fmaf(hj, w1r[hh + 3], c[hh + 3]);
      acc0 = fmaf(a4r[hh + 0], fabsf(p0), acc0);
      acc1 = fmaf(a4r[hh + 1], fabsf(p1), acc1);
      acc2 = fmaf(a4r[hh + 2], fabsf(p2), acc2);
      acc3 = fmaf(a4r[hh + 3], fabsf(p3), acc3);
    }
    et[jj][k] = 0.6f * fmaf(hj, S1, Cdot) + ((acc0 + acc1) + (acc2 + acc3));
  }
  __syncthreads();

  // softmax over k (256) per row jj; wave32: each wave does rows wv, wv+8
  const int wv = tid >> 5;
  const int ln = tid & 31;
  for (int jj = wv; jj < 16; jj += 8) {
    float v[8];
    float m = -INFINITY;
#pragma unroll
    for (int i = 0; i < 8; ++i) { v[i] = et[jj][ln + 32 * i]; m = fmaxf(m, v[i]); }
#pragma unroll
    for (int off = 16; off > 0; off >>= 1) m = fmaxf(m, __shfl_xor(m, off, 32));
    float s = 0.0f;
#pragma unroll
    for (int i = 0; i < 8; ++i) { v[i] = __expf(v[i] - m); s += v[i]; }
#pragma unroll
    for (int off = 16; off > 0; off >>= 1) s += __shfl_xor(s, off, 32);
    const float rs = 1.0f / s;
    const size_t obase = ((size_t)(b * N_ + jt * 16 + jj)) * N_ * W_ + t;
#pragma unroll
    for (int i = 0; i < 8; ++i) {
      const int kk = ln + 32 * i;
      alpha[obase + (size_t)kk * W_] = v[i] * rs;
    }
  }
}

// ---------------------------------------------------------------------------
// Kernel 3: scalar GRU over t for each (b,j,k) row, then softplus -> E
// ---------------------------------------------------------------------------
__global__ __launch_bounds__(256) void gru_kernel(
    const float* __restrict__ alpha,  // rows of 64 (t contiguous)
    const float* __restrict__ wih, const float* __restrict__ whh,
    const float* __restrict__ bih, const float* __restrict__ bhh,
    float* __restrict__ E)            // (8*256*256,)
{
  const int row = blockIdx.x * 256 + threadIdx.x;   // 0..524287
  const float wi0 = wih[0], wi1 = wih[1], wi2 = wih[2];
  const float wh0 = whh[0], wh1 = whh[1], wh2 = whh[2];
  const float bi0 = bih[0], bi1 = bih[1], bi2 = bih[2];
  const float bh0 = bhh[0], bh1 = bhh[1], bh2 = bhh[2];

  const float4* __restrict__ arow = (const float4*)(alpha + (size_t)row * 64);
  __builtin_prefetch(arow, 0, 0);
  float hc = 0.0f;
  for (int c4 = 0; c4 < 16; ++c4) {
    const float4 x4 = arow[c4];
    const float xs[4] = {x4.x, x4.y, x4.z, x4.w};
#pragma unroll
    for (int i = 0; i < 4; ++i) {
      const float x = xs[i];
      const float gi0 = fmaf(x, wi0, bi0);
      const float gi1 = fmaf(x, wi1, bi1);
      const float gi2 = fmaf(x, wi2, bi2);
      const float gh0 = fmaf(hc, wh0, bh0);
      const float gh1 = fmaf(hc, wh1, bh1);
      const float gh2 = fmaf(hc, wh2, bh2);
      const float r = 1.0f / (1.0f + __expf(-(gi0 + gh0)));
      const float z = 1.0f / (1.0f + __expf(-(gi1 + gh1)));
      const float nn = tanhf(fmaf(r, gh2, gi2));
      hc = fmaf(z, hc - nn, nn);    // (1-z)*nn + z*hc
    }
  }
  // stable softplus
  E[row] = fmaxf(hc, 0.0f) + log1pf(__expf(-fabsf(hc)));
}

// ---------------------------------------------------------------------------
// Kernel 4: edge index (as float) + symmetrized edge weights
// ---------------------------------------------------------------------------
__global__ __launch_bounds__(256) void edge_kernel(
    const float* __restrict__ E, float* __restrict__ out)
{
  const int idx = blockIdx.x * 256 + threadIdx.x;
  if (idx >= B_ * PAIRS) return;
  const int b = idx / PAIRS;
  const int p = idx % PAIRS;
  const int s = p / 255;
  const int r = p % 255;
  const int d = r + (r >= s ? 1 : 0);
  out[idx]                = (float)(s + b * 256);          // edge_index row 0
  out[522240 + idx]       = (float)(d + b * 256);          // edge_index row 1
  const size_t eb = (size_t)b * 65536;
  out[1044480 + idx] = 0.5f * (E[eb + s * 256 + d] + E[eb + d * 256 + s]);
}

// ---------------------------------------------------------------------------
extern "C" void kernel_launch(void* const* d_in, const int* in_sizes, int n_in,
                              void* d_out, int out_size, void* d_ws, size_t ws_size,
                              hipStream_t stream) {
  const float* h   = (const float*)d_in[0];
  const float* Ww  = (const float*)d_in[1];
  const float* Wb  = (const float*)d_in[2];
  const float* att = (const float*)d_in[3];
  const float* wih = (const float*)d_in[4];
  const float* whh = (const float*)d_in[5];
  const float* bih = (const float*)d_in[6];
  const float* bhh = (const float*)d_in[7];
  float* out = (float*)d_out;
  float* ws  = (float*)d_ws;

  float* E     = ws;             // 524288 floats
  float* base  = ws + 524288;    // 4096 floats
  float* alpha = out + OUT_ALPHA_OFF;

  base_wmma_kernel<<<1, 512, 0, stream>>>(Ww, Wb, base);
  dim3 grid(16, 64, 8);
  attn_kernel<<<grid, 256, 0, stream>>>(h, Ww, att, base, alpha);
  gru_kernel<<<2048, 256, 0, stream>>>(alpha, wih, whh, bih, bhh, E);
  edge_kernel<<<2040, 256, 0, stream>>>(E, out);
}